// GraphConvolution_38852274160228
// MI455X (gfx1250) — compile-verified
//
#include <hip/hip_runtime.h>
#include <hip/hip_bf16.h>

#define N_NODES 100000
#define N_EDGES 1600000
#define DIN 128
#define DOUT 128
#define N_SUP 2

typedef __attribute__((ext_vector_type(2))) float v2f;
typedef __attribute__((ext_vector_type(8))) float v8f;

// ---------------------------------------------------------------------------
// Phase 1: pre[s] = x @ w[s]   using V_WMMA_F32_16X16X4_F32
// grid = (N_NODES/16, N_SUP), block = 256 (8 waves; wave i -> N-tile i)
// ---------------------------------------------------------------------------
__global__ __launch_bounds__(256) void gemm_pre_kernel(
    const float* __restrict__ x,   // [N_NODES, DIN]
    const float* __restrict__ w,   // [N_SUP, DIN, DOUT]
    float* __restrict__ pre)       // [N_SUP, N_NODES, DOUT]
{
    const int s    = blockIdx.y;
    const int mt   = blockIdx.x;            // M tile index (16 rows)
    const int wave = threadIdx.x >> 5;      // N tile index 0..7
    const int lane = threadIdx.x & 31;

    const int m0   = mt * 16;
    const int n0   = wave * 16;
    const int row  = lane & 15;             // A-matrix M for this lane
    const int koff = (lane >> 4) * 2;       // 0 for lanes 0-15, 2 for 16-31
    const int n    = lane & 15;             // B/C/D N for this lane

    const float* xa = x + (size_t)(m0 + row) * DIN;
    const float* wb = w + (size_t)s * DIN * DOUT + n0 + n;

    v8f c = {};
    #pragma unroll 4
    for (int k0 = 0; k0 < DIN; k0 += 4) {
        // A fragment: 16x4 fp32. VGPR0 = K=k0 (lanes 0-15) / K=k0+2 (16-31),
        //             VGPR1 = K=k0+1 / K=k0+3.
        v2f a, b;
        a.x = xa[k0 + koff + 0];
        a.y = xa[k0 + koff + 1];
        // B fragment: 4x16 fp32, same K split across lane halves, N = lane&15.
        b.x = wb[(size_t)(k0 + koff + 0) * DOUT];
        b.y = wb[(size_t)(k0 + koff + 1) * DOUT];
        c = __builtin_amdgcn_wmma_f32_16x16x4_f32(
                /*neg_a=*/false, a, /*neg_b=*/false, b,
                /*c_mod=*/(short)0, c, /*reuse_a=*/false, /*reuse_b=*/false);
    }

    // C/D layout: VGPR r -> M = r (lanes 0-15) / r+8 (lanes 16-31), N = lane&15
    float* op = pre + (size_t)s * N_NODES * DOUT;
    const int mbase = m0 + (lane >> 4) * 8;
    #pragma unroll
    for (int r = 0; r < 8; ++r) {
        op[(size_t)(mbase + r) * DOUT + n0 + n] = c[r];
    }
}

// ---------------------------------------------------------------------------
// Phase 2: zero the output accumulator
// ---------------------------------------------------------------------------
__global__ __launch_bounds__(256) void zero_out_kernel(float4* __restrict__ out, int n4)
{
    int i = blockIdx.x * blockDim.x + threadIdx.x;
    if (i < n4) out[i] = make_float4(0.f, 0.f, 0.f, 0.f);
}

// ---------------------------------------------------------------------------
// Phase 3: COO scatter  out[row] += val * pre[s][col]
// 32 lanes per edge, 4 floats per lane. grid = (E/8, N_SUP), block = 256.
// ---------------------------------------------------------------------------
__global__ __launch_bounds__(256) void spmm_scatter_kernel(
    const float* __restrict__ pre,        // [N_SUP, N_NODES, DOUT]
    const float* __restrict__ sup_vals,   // [N_SUP, N_EDGES]
    const int*   __restrict__ sup_rows,   // [N_SUP, N_EDGES]
    const int*   __restrict__ sup_cols,   // [N_SUP, N_EDGES]
    float* __restrict__ out)              // [N_NODES, DOUT]
{
    const int s  = blockIdx.y;
    const long long e = (long long)blockIdx.x * 8 + (threadIdx.x >> 5);
    if (e >= N_EDGES) return;
    const int lane = threadIdx.x & 31;

    const size_t eb = (size_t)s * N_EDGES + (size_t)e;
    const float v = sup_vals[eb];
    const int   r = sup_rows[eb];
    const int   c = sup_cols[eb];

    const float4 p = *(const float4*)(pre + (size_t)s * N_NODES * DOUT
                                          + (size_t)c * DOUT + lane * 4);
    float* o = out + (size_t)r * DOUT + lane * 4;
    atomicAdd(o + 0, v * p.x);
    atomicAdd(o + 1, v * p.y);
    atomicAdd(o + 2, v * p.z);
    atomicAdd(o + 3, v * p.w);
}

// ---------------------------------------------------------------------------
// Phase 4: out = relu(out + bias)
// ---------------------------------------------------------------------------
__global__ __launch_bounds__(256) void bias_relu_kernel(
    float4* __restrict__ out, const float* __restrict__ bias, int n4)
{
    int i = blockIdx.x * blockDim.x + threadIdx.x;
    if (i >= n4) return;
    const int c4 = (i & 31) * 4;   // DOUT/4 == 32 float4 per row
    float4 t = out[i];
    t.x = fmaxf(t.x + bias[c4 + 0], 0.f);
    t.y = fmaxf(t.y + bias[c4 + 1], 0.f);
    t.z = fmaxf(t.z + bias[c4 + 2], 0.f);
    t.w = fmaxf(t.w + bias[c4 + 3], 0.f);
    out[i] = t;
}

// ---------------------------------------------------------------------------
extern "C" void kernel_launch(void* const* d_in, const int* in_sizes, int n_in,
                              void* d_out, int out_size, void* d_ws, size_t ws_size,
                              hipStream_t stream)
{
    const float* x        = (const float*)d_in[0];
    const float* w        = (const float*)d_in[1];
    const float* bias     = (const float*)d_in[2];
    const float* sup_vals = (const float*)d_in[3];
    const int*   sup_rows = (const int*)d_in[4];
    const int*   sup_cols = (const int*)d_in[5];
    float* out = (float*)d_out;
    float* pre = (float*)d_ws;   // needs N_SUP*N_NODES*DOUT*4 = 102.4 MB

    // Phase 1: dense GEMMs via fp32 WMMA
    dim3 g1(N_NODES / 16, N_SUP);
    gemm_pre_kernel<<<g1, 256, 0, stream>>>(x, w, pre);

    // Phase 2: zero accumulator
    const int n4 = (N_NODES * DOUT) / 4;
    zero_out_kernel<<<(n4 + 255) / 256, 256, 0, stream>>>((float4*)out, n4);

    // Phase 3: edge scatter with f32 atomics (L2-resident working set)
    dim3 g3((N_EDGES + 7) / 8, N_SUP);
    spmm_scatter_kernel<<<g3, 256, 0, stream>>>(pre, sup_vals, sup_rows, sup_cols, out);

    // Phase 4: bias + relu
    bias_relu_kernel<<<(n4 + 255) / 256, 256, 0, stream>>>((float4*)out, bias, n4);
}